// NegativeLearningLossRandomSample_1116691497360
// MI455X (gfx1250) — compile-verified
//
#include <hip/hip_runtime.h>
#include <math.h>

// Problem constants (from the reference)
#define B_      4
#define S_      1024
#define V_      32000
#define VW_     (V_ / 32)      // 1000 mask words per batch
#define POOL_   1024           // TOKENS_NUM * TOKEN_FACTOR
#define TOK_    256            // TOKENS_NUM
#define BLOCK_  256            // 8 waves (wave32)
#define CAP_    2048           // pool capacity incl. last-byte ties
#define NROWS_  (B_ * S_)

// ---- gfx1250 async global->LDS path (probe via __has_builtin; fallback = sync copy)
#if defined(__HIP_DEVICE_COMPILE__) && defined(__has_builtin)
#if __has_builtin(__builtin_amdgcn_global_load_async_to_lds_b128)
#define HAVE_ASYNC_LDS 1
#endif
#if __has_builtin(__builtin_amdgcn_s_wait_asynccnt)
#define HAVE_WAIT_ASYNC 1
#endif
#endif

#if defined(HAVE_ASYNC_LDS)
typedef int v4i_ __attribute__((ext_vector_type(4)));
typedef __attribute__((address_space(1))) v4i_* gv4p_;   // global int4*
typedef __attribute__((address_space(3))) v4i_* lv4p_;   // LDS int4*
#endif

// Monotone float->uint key (descending float order == descending uint order)
__device__ __forceinline__ unsigned fkey(float x) {
    unsigned u = __float_as_uint(x);
    return (u & 0x80000000u) ? ~u : (u | 0x80000000u);
}

__device__ __forceinline__ unsigned hash32(unsigned x) {
    x ^= x >> 16; x *= 0x7feb352dU;
    x ^= x >> 15; x *= 0x846ca68bU;
    x ^= x >> 16;
    return x;
}

// ---------------- kernel 0: zero the per-batch vocab bitmask ----------------
__global__ void zero_mask_kernel(unsigned* __restrict__ mask, int n) {
    int i = blockIdx.x * blockDim.x + threadIdx.x;
    if (i < n) mask[i] = 0u;
}

// ---------------- kernel 1: scatter target bits -----------------------------
__global__ void scatter_mask_kernel(const int* __restrict__ targets,
                                    unsigned* __restrict__ mask) {
    int i = blockIdx.x * blockDim.x + threadIdx.x;
    if (i < NROWS_) {
        int b = i / S_;
        unsigned t = (unsigned)targets[i];
        if (t < (unsigned)V_)
            atomicOr(&mask[b * VW_ + (t >> 5)], 1u << (t & 31u));
    }
}

// ---------------- kernel 2: per-row softmax + top-1024 + sample-256 ---------
__global__ void __launch_bounds__(BLOCK_)
row_kernel(const float* __restrict__ inputs,
           const unsigned* __restrict__ mask,
           float* __restrict__ partials) {
    extern __shared__ char smem[];
    float*    row   = (float*)smem;                 // 32000 f
    float*    poolp = row + V_;                     // 2048 f
    unsigned* poolh = (unsigned*)(poolp + CAP_);    // 2048 u
    unsigned* hist  = poolh + CAP_;                 // 256 u
    float*    red   = (float*)(hist + 256);         // 256 f
    unsigned* scal  = (unsigned*)(red + 256);       // 16 u
    unsigned* lmask = scal + 16;                    // 1000 u

    const int tid   = threadIdx.x;
    const int rowId = blockIdx.x;
    const int b     = rowId / S_;
    const float* rp = inputs + (size_t)rowId * V_;

    // batch mask -> LDS (4 KB, L2-hot)
    for (int i = tid; i < VW_; i += BLOCK_) lmask[i] = mask[b * VW_ + i];

    // Stream the 125 KB row into LDS exactly once.
#if defined(HAVE_ASYNC_LDS)
    {
        const char* gsrc = (const char*)rp;
        char*       dstc = (char*)row;
        for (int i = tid; i < V_ / 4; i += BLOCK_) {   // 32000 dwords = 8000 x b128
            __builtin_amdgcn_global_load_async_to_lds_b128(
                (gv4p_)(gsrc + (size_t)i * 16),
                (lv4p_)(dstc + (size_t)i * 16),
                0, 0);
        }
#if defined(HAVE_WAIT_ASYNC)
        __builtin_amdgcn_s_wait_asynccnt(0);
#else
        asm volatile("s_wait_asynccnt 0" ::: "memory");
#endif
    }
#else
    {
        const float4* g4 = (const float4*)rp;
        float4*       s4 = (float4*)row;
        for (int i = tid; i < V_ / 4; i += BLOCK_) s4[i] = g4[i];
    }
#endif
    __syncthreads();

    // ---- pass 1: row max (unmasked; softmax uses raw logits) ----
    float lm = -INFINITY;
    for (int v = tid; v < V_; v += BLOCK_) lm = fmaxf(lm, row[v]);
    red[tid] = lm; __syncthreads();
    for (int off = BLOCK_ / 2; off > 0; off >>= 1) {
        if (tid < off) red[tid] = fmaxf(red[tid], red[tid + off]);
        __syncthreads();
    }
    const float m = red[0]; __syncthreads();

    // ---- pass 2: sum exp ----
    float ls = 0.f;
    for (int v = tid; v < V_; v += BLOCK_) ls += __expf(row[v] - m);
    red[tid] = ls; __syncthreads();
    for (int off = BLOCK_ / 2; off > 0; off >>= 1) {
        if (tid < off) red[tid] += red[tid + off];
        __syncthreads();
    }
    const float invZ = 1.f / red[0]; __syncthreads();

    // ---- radix-select the top-POOL_ threshold over masked keys (3x8 bits) ----
    unsigned prefix = 0, himask = 0, want = POOL_;
    for (int level = 0; level < 3; ++level) {
        const int shift = 24 - 8 * level;
        if (tid < 256) hist[tid] = 0u;
        __syncthreads();
        for (int v = tid; v < V_; v += BLOCK_) {
            unsigned bit = (lmask[v >> 5] >> (v & 31u)) & 1u;
            unsigned key = bit ? 0u : fkey(row[v]);
            if ((key & himask) == prefix)
                atomicAdd(&hist[(key >> shift) & 0xFFu], 1u);
        }
        __syncthreads();
        if (tid == 0) {
            unsigned acc = 0; int bin = 255;
            for (; bin > 0; --bin) {
                unsigned h = hist[bin];
                if (acc + h >= want) break;
                acc += h;
            }
            scal[0] = (unsigned)bin;
            scal[1] = want - acc;
        }
        __syncthreads();
        prefix |= scal[0] << shift;
        himask |= 0xFFu << shift;
        want    = scal[1];
        __syncthreads();
    }
    const unsigned tau = prefix;   // low byte 0: keys >= tau form the pool (>= POOL_ entries)

    // ---- gather pool: probability + deterministic per-token noise ----
    if (tid == 0) scal[2] = 0u;
    __syncthreads();
    for (int v = tid; v < V_; v += BLOCK_) {
        unsigned bit = (lmask[v >> 5] >> (v & 31u)) & 1u;
        unsigned key = bit ? 0u : fkey(row[v]);
        if (key >= tau && key != 0u) {
            unsigned pos = atomicAdd(&scal[2], 1u);
            if (pos < CAP_) {
                poolp[pos] = __expf(row[v] - m) * invZ;
                poolh[pos] = hash32((unsigned)v * 0x9E3779B1u +
                                    (unsigned)rowId * 0x85EBCA77u + 0x165667B1u);
            }
        }
    }
    __syncthreads();
    const unsigned P = scal[2] < CAP_ ? scal[2] : CAP_;

    // ---- radix-select the noise threshold for exactly TOK_ picks (4x8 bits) ----
    unsigned npref = 0;
    if (P > TOK_) {
        unsigned nhi = 0, nwant = TOK_;
        for (int level = 0; level < 4; ++level) {
            const int shift = 24 - 8 * level;
            if (tid < 256) hist[tid] = 0u;
            __syncthreads();
            for (unsigned i = tid; i < P; i += BLOCK_) {
                unsigned k = poolh[i];
                if ((k & nhi) == npref)
                    atomicAdd(&hist[(k >> shift) & 0xFFu], 1u);
            }
            __syncthreads();
            if (tid == 0) {
                unsigned acc = 0; int bin = 255;
                for (; bin > 0; --bin) {
                    unsigned h = hist[bin];
                    if (acc + h >= nwant) break;
                    acc += h;
                }
                scal[3] = (unsigned)bin;
                scal[4] = nwant - acc;
            }
            __syncthreads();
            npref |= scal[3] << shift;
            nhi   |= 0xFFu << shift;
            nwant  = scal[4];
            __syncthreads();
        }
    }

    // ---- accumulate -log(1 - p) over selected pool entries ----
    float acc = 0.f;
    for (unsigned i = tid; i < P; i += BLOCK_) {
        if (poolh[i] >= npref) {
            float p = fminf(poolp[i], 0.99999988f);
            acc -= log1pf(-p);
        }
    }
    red[tid] = acc; __syncthreads();
    for (int off = BLOCK_ / 2; off > 0; off >>= 1) {
        if (tid < off) red[tid] += red[tid + off];
        __syncthreads();
    }
    if (tid == 0) partials[rowId] = red[0];
}

// ---------------- kernel 3: deterministic fixed-order final reduction -------
__global__ void final_reduce_kernel(const float* __restrict__ partials,
                                    float* __restrict__ out) {
    __shared__ float sr[BLOCK_];
    float a = 0.f;
    for (int i = threadIdx.x; i < NROWS_; i += BLOCK_) a += partials[i];
    sr[threadIdx.x] = a; __syncthreads();
    for (int off = BLOCK_ / 2; off > 0; off >>= 1) {
        if (threadIdx.x < off) sr[threadIdx.x] += sr[threadIdx.x + off];
        __syncthreads();
    }
    if (threadIdx.x == 0) out[0] = sr[0];
}

extern "C" void kernel_launch(void* const* d_in, const int* in_sizes, int n_in,
                              void* d_out, int out_size, void* d_ws, size_t ws_size,
                              hipStream_t stream) {
    (void)in_sizes; (void)n_in; (void)out_size; (void)ws_size;
    const float* inputs  = (const float*)d_in[0];
    const int*   targets = (const int*)d_in[1];   // jnp int64 -> int32 under default JAX config

    unsigned* mask     = (unsigned*)d_ws;                     // 4000 u32 = 16000 B
    float*    partials = (float*)((char*)d_ws + 16384);       // 4096 f32

    constexpr size_t SMEM_BYTES =
        (size_t)(V_ + CAP_ + 256) * sizeof(float) +
        (size_t)(CAP_ + 256 + 16 + VW_) * sizeof(unsigned);   // ~150.5 KB (fits 320 KB WGP LDS)

    // allow >64KB dynamic LDS where the runtime wants the attribute; ignore errors
    (void)hipFuncSetAttribute((const void*)row_kernel,
                              hipFuncAttributeMaxDynamicSharedMemorySize,
                              (int)SMEM_BYTES);

    zero_mask_kernel<<<(B_ * VW_ + 255) / 256, 256, 0, stream>>>(mask, B_ * VW_);
    scatter_mask_kernel<<<(NROWS_ + 255) / 256, 256, 0, stream>>>(targets, mask);
    row_kernel<<<NROWS_, BLOCK_, SMEM_BYTES, stream>>>(inputs, mask, partials);
    final_reduce_kernel<<<1, BLOCK_, 0, stream>>>(partials, (float*)d_out);
}